// HybridForecaster_14431090114664
// MI455X (gfx1250) — compile-verified
//
#include <hip/hip_runtime.h>
#include <math.h>
#include <stdint.h>

// ---------------------------------------------------------------------------
// QG 3-layer ensemble forecast for MI455X (gfx1250, wave32).
// DST-I elliptic solve implemented as batched 512^3 FP32 GEMMs on the WMMA
// pipe (v_wmma_f32_16x16x4_f32), with CDNA5 async global->LDS staging
// (global_load_async_to_lds_b128 / s_wait_asynccnt) and double-buffered
// LDS panels so copy latency hides behind the WMMA stream.
// ---------------------------------------------------------------------------

typedef float v2f __attribute__((ext_vector_type(2)));
typedef float v8f __attribute__((ext_vector_type(8)));

constexpr int    NXg   = 513;
constexpr int    NIg   = 511;
constexpr int    NPg   = 512;
constexpr int    Eg    = 4;     // ensembles
constexpr int    Lg    = 3;     // layers
constexpr size_t FIELD = (size_t)NXg * NXg;        // 263169
constexpr size_t ELXY  = (size_t)Eg * Lg * FIELD;  // 12 fields
constexpr size_t STATE = 2 * ELXY;                 // (q,p)
constexpr size_t NPNP  = (size_t)NPg * NPg;
constexpr size_t GSZ   = (size_t)Eg * Lg * NPNP;

// physics constants (float64 host math folded into f32 literals)
#define ZFBCc  ((float)(0.2 / 1.1))
#define JACCc  ((float)(1.0 / (9.375e-05 * 1.0e8)))                 // 1/(F0*DX*DY)
#define HDCc   ((float)(2.0e9 / (9.375e-05 * 9.375e-05) / 1.0e24))  // A4/F0^2/DX^6
#define BFCc   ((float)(2.0 / (2.0 * 9.375e-05 * 1.0e8 * (-2900.0))))
#define BNDCc  ((float)((0.2 / 1.1) / (9.375e-05 * 1.0e4 * 9.375e-05 * 1.0e4)))
#define PI_F   3.14159265358979323846f

// ---------------------------------------------------------------------------
// CDNA5 async global -> LDS copy, 16B per lane, tracked by ASYNCcnt.
// Flat shared-aperture addresses carry the LDS byte offset in their low 32
// bits (ISA 10.2 per-aperture address calc), so truncation yields VDST.
// ---------------------------------------------------------------------------
__device__ __forceinline__ void async_copy16(const float* g, const float* lds) {
  unsigned           l  = (unsigned)(uintptr_t)lds;
  unsigned long long ga = (unsigned long long)(uintptr_t)g;
  asm volatile("global_load_async_to_lds_b128 %0, %1, off"
               :: "v"(l), "v"(ga) : "memory");
}
__device__ __forceinline__ void wait_async0() {
  asm volatile("s_wait_asynccnt 0x0" ::: "memory");
}

// ---------------------------------------------------------------------------
// Orthonormal DST-I matrix, padded to 512x512. sqrt(2/512) = 1/16 exactly.
// Angle reduced mod 2*pi via integer mod 1024 so sinf stays accurate.
// ---------------------------------------------------------------------------
__global__ void make_S_kernel(float* __restrict__ S) {
  int i = blockIdx.y * 16 + threadIdx.y;
  int j = blockIdx.x * 16 + threadIdx.x;
  float v = 0.0f;
  if (i < NIg && j < NIg) {
    int r = ((i + 1) * (j + 1)) & 1023;            // period 1024 <-> 2*pi
    v = 0.0625f * __sinf(PI_F * (float)r * (1.0f / 512.0f));
  }
  S[(size_t)i * NPg + j] = v;
}

// ---------------------------------------------------------------------------
// Batched 512x512x512 FP32 GEMM: C[z] = A[z] * B[z] (stride 0 => shared op).
// 256 threads = 8 waves; block tile 64(M) x 64(N); each wave owns a 16x32
// slab of C (two v8f accumulators, A fragment reused across both WMMAs).
// Double-buffered LDS panels staged with global_load_async_to_lds_b128:
// panel k+16 streams in while panel k feeds the WMMA pipe; one barrier
// per K-panel. Fragment layouts per CDNA5 ISA 7.12.2.
// ---------------------------------------------------------------------------
__global__ void __launch_bounds__(256)
gemm512_wmma(const float* __restrict__ A, long long aStride,
             const float* __restrict__ B, long long bStride,
             float* __restrict__ C) {
  const int bz = blockIdx.z;
  A += (long long)bz * aStride;
  B += (long long)bz * bStride;
  C += (size_t)bz * NPNP;

  __shared__ float As[2][64][20];   // 80B row stride: 16B aligned, conflict-free
  __shared__ float Bs[2][16][68];   // 272B row stride: 16B aligned

  const int tid  = threadIdx.x;
  const int wave = tid >> 5;
  const int lane = tid & 31;
  const int half = lane >> 4;          // K-half (A,B), M-half (C)
  const int ln16 = lane & 15;          // M index (A), N index (B,C)
  const int wm   = wave >> 1;          // 0..3 (M tile)
  const int wn   = wave & 1;           // 0..1 (pair of N tiles)
  const int row0 = blockIdx.y * 64;
  const int col0 = blockIdx.x * 64;

  // per-thread 16B staging chunk coordinates
  const int ar = tid >> 2, aq = (tid & 3) << 2;    // A: 64 rows x 4 chunks
  const int br = tid >> 4, bq = (tid & 15) << 2;   // B: 16 rows x 16 chunks

  v8f acc0 = {}, acc1 = {};

  // prologue: stage panel 0 into buffer 0
  async_copy16(&A[(size_t)(row0 + ar) * NPg + aq], &As[0][ar][aq]);
  async_copy16(&B[(size_t)br * NPg + col0 + bq], &Bs[0][br][bq]);
  wait_async0();
  __syncthreads();

  int buf = 0;
  for (int kb = 0; kb < NPg; kb += 16) {
    const int nbuf = buf ^ 1;
    if (kb + 16 < NPg) {   // stream next panel while this one computes
      async_copy16(&A[(size_t)(row0 + ar) * NPg + kb + 16 + aq], &As[nbuf][ar][aq]);
      async_copy16(&B[(size_t)(kb + 16 + br) * NPg + col0 + bq], &Bs[nbuf][br][bq]);
    }

#pragma unroll
    for (int kk = 0; kk < 16; kk += 4) {
      v2f a, b0, b1;
      // A 16x4: lanes 0-15 K={kk,kk+1}, lanes 16-31 K={kk+2,kk+3}
      a.x  = As[buf][wm * 16 + ln16][kk + 2 * half + 0];
      a.y  = As[buf][wm * 16 + ln16][kk + 2 * half + 1];
      // B 4x16: row striped across lanes; rows split across lane halves
      b0.x = Bs[buf][kk + 2 * half + 0][wn * 32 + ln16];
      b0.y = Bs[buf][kk + 2 * half + 1][wn * 32 + ln16];
      b1.x = Bs[buf][kk + 2 * half + 0][wn * 32 + 16 + ln16];
      b1.y = Bs[buf][kk + 2 * half + 1][wn * 32 + 16 + ln16];
      acc0 = __builtin_amdgcn_wmma_f32_16x16x4_f32(
          false, a, false, b0, (short)0, acc0, false, false);
      acc1 = __builtin_amdgcn_wmma_f32_16x16x4_f32(
          false, a, false, b1, (short)0, acc1, false, false);
    }

    wait_async0();     // this wave's next-panel copies have landed in LDS
    __syncthreads();   // all waves done reading `buf`, all copies visible
    buf = nbuf;
  }

  // C/D layout: VGPR v -> M = v + 8*half, N = ln16
#pragma unroll
  for (int v = 0; v < 8; ++v) {
    int r = row0 + wm * 16 + v + 8 * half;
    C[(size_t)r * NPg + col0 + wn * 32 + ln16]      = acc0[v];
    C[(size_t)r * NPg + col0 + wn * 32 + 16 + ln16] = acc1[v];
  }
}

// ---------------------------------------------------------------------------
// laplacian_h with free-slip BC factor ZFBC (col rules override corners)
// ---------------------------------------------------------------------------
__global__ void lap_bc_kernel(const float* __restrict__ f, float* __restrict__ out) {
  int m = blockIdx.z;
  const float* F = f + (size_t)m * FIELD;
  float* O = out + (size_t)m * FIELD;
  int x = blockIdx.y * 16 + threadIdx.y;
  int y = blockIdx.x * 16 + threadIdx.x;
  if (x >= NXg || y >= NXg) return;
  float r;
  if (y == 0)            r = ZFBCc * (F[(size_t)x * NXg + 1] - F[(size_t)x * NXg + 0]);
  else if (y == NXg - 1) r = ZFBCc * (F[(size_t)x * NXg + NXg - 2] - F[(size_t)x * NXg + NXg - 1]);
  else if (x == 0)       r = ZFBCc * (F[(size_t)1 * NXg + y] - F[y]);
  else if (x == NXg - 1) r = ZFBCc * (F[(size_t)(NXg - 2) * NXg + y] - F[(size_t)(NXg - 1) * NXg + y]);
  else
    r = F[(size_t)(x + 1) * NXg + y] + F[(size_t)(x - 1) * NXg + y] +
        F[(size_t)x * NXg + y + 1] + F[(size_t)x * NXg + y - 1] -
        4.0f * F[(size_t)x * NXg + y];
  O[(size_t)x * NXg + y] = r;
}

// ---------------------------------------------------------------------------
// rhs interior: Arakawa Jacobian + hyperdiffusion + wind + bottom friction
// ---------------------------------------------------------------------------
__global__ void rhs_kernel(const float* __restrict__ Y, const float* __restrict__ LP2,
                           const float* __restrict__ wind, float* __restrict__ dq) {
  int m = blockIdx.z;          // e*3 + l
  int l = m % Lg;
  int x = blockIdx.y * 16 + threadIdx.y;
  int y = blockIdx.x * 16 + threadIdx.x;
  if (x >= NXg || y >= NXg) return;
  size_t base = (size_t)m * FIELD;
  size_t idx  = (size_t)x * NXg + y;
  if (x == 0 || y == 0 || x == NXg - 1 || y == NXg - 1) { dq[base + idx] = 0.0f; return; }

  const float* q = Y + base;
  const float* p = Y + ELXY + base;
#define QQ(i, j) q[(size_t)(i) * NXg + (j)]
#define PP(i, j) p[(size_t)(i) * NXg + (j)]
  float dxf = QQ(x + 1, y) - QQ(x - 1, y), dyf = QQ(x, y + 1) - QQ(x, y - 1);
  float dxg = PP(x + 1, y) - PP(x - 1, y), dyg = PP(x, y + 1) - PP(x, y - 1);
  float jac =
      dxf * dyg - dxg * dyf
    + (QQ(x + 1, y) * (PP(x + 1, y + 1) - PP(x + 1, y - 1)) -
       QQ(x - 1, y) * (PP(x - 1, y + 1) - PP(x - 1, y - 1)))
    - (QQ(x, y + 1) * (PP(x + 1, y + 1) - PP(x - 1, y + 1)) -
       QQ(x, y - 1) * (PP(x + 1, y - 1) - PP(x - 1, y - 1)))
    + (PP(x, y + 1) * (QQ(x + 1, y + 1) - QQ(x - 1, y + 1)) -
       PP(x, y - 1) * (QQ(x + 1, y - 1) - QQ(x - 1, y - 1)))
    - (PP(x + 1, y) * (QQ(x + 1, y + 1) - QQ(x + 1, y - 1)) -
       PP(x - 1, y) * (QQ(x - 1, y + 1) - QQ(x - 1, y - 1)));
  jac *= (1.0f / 12.0f);

  const float* L2 = LP2 + base;
  float lap4 = L2[(size_t)(x + 1) * NXg + y] + L2[(size_t)(x - 1) * NXg + y] +
               L2[(size_t)x * NXg + y + 1] + L2[(size_t)x * NXg + y - 1] -
               4.0f * L2[(size_t)x * NXg + y];

  float r = JACCc * jac - HDCc * lap4;
  if (l == 0) r += wind[(size_t)(x - 1) * NIg + (y - 1)];
  if (l == Lg - 1) {
    float lapP = PP(x + 1, y) + PP(x - 1, y) + PP(x, y + 1) + PP(x, y - 1) - 4.0f * PP(x, y);
    r += BFCc * lapP;
  }
  dq[base + idx] = r;
#undef QQ
#undef PP
}

// layer->mode transform fused with pack into padded 512x512 GEMM buffer
__global__ void cl2m_pack_kernel(const float* __restrict__ dq, const float* __restrict__ Cl2m,
                                 float* __restrict__ GA) {
  int e  = blockIdx.z;
  int ii = blockIdx.y * 16 + threadIdx.y;   // 0..511
  int jj = blockIdx.x * 16 + threadIdx.x;
  float v0 = 0.0f, v1 = 0.0f, v2 = 0.0f;
  if (ii < NIg && jj < NIg) {
    size_t idx = (size_t)(ii + 1) * NXg + (jj + 1);
    float d0 = dq[((size_t)(e * Lg + 0)) * FIELD + idx];
    float d1 = dq[((size_t)(e * Lg + 1)) * FIELD + idx];
    float d2 = dq[((size_t)(e * Lg + 2)) * FIELD + idx];
    v0 = Cl2m[0] * d0 + Cl2m[1] * d1 + Cl2m[2] * d2;
    v1 = Cl2m[3] * d0 + Cl2m[4] * d1 + Cl2m[5] * d2;
    v2 = Cl2m[6] * d0 + Cl2m[7] * d1 + Cl2m[8] * d2;
  }
  size_t g = (size_t)ii * NPg + jj;
  GA[((size_t)(e * Lg + 0)) * NPNP + g] = v0;
  GA[((size_t)(e * Lg + 1)) * NPNP + g] = v1;
  GA[((size_t)(e * Lg + 2)) * NPNP + g] = v2;
}

// spectral divide by Helmholtz operator
__global__ void helm_scale_kernel(float* __restrict__ GA, const float* __restrict__ helm) {
  int m = blockIdx.z;
  int mode = m % Lg;
  int ii = blockIdx.y * 16 + threadIdx.y;
  int jj = blockIdx.x * 16 + threadIdx.x;
  if (ii < NIg && jj < NIg) {
    size_t g = (size_t)m * NPNP + (size_t)ii * NPg + jj;
    GA[g] /= helm[(size_t)mode * NIg * NIg + (size_t)ii * NIg + jj];
  }
}

// mean of dp_modes (interior sum / 513^2) for modes 0..1
__global__ void mean_reduce_kernel(const float* __restrict__ GA, float* __restrict__ MEAN) {
  int b = blockIdx.x;                 // 0..7 : e*2 + j
  int e = b >> 1, j = b & 1;
  const float* G = GA + ((size_t)(e * Lg + j)) * NPNP;
  __shared__ float sm[256];
  float s = 0.0f;
  for (int idx = threadIdx.x; idx < NIg * NIg; idx += 256) {
    int i = idx / NIg, jj = idx - i * NIg;
    s += G[(size_t)i * NPg + jj];
  }
  sm[threadIdx.x] = s;
  __syncthreads();
  for (int o = 128; o > 0; o >>= 1) {
    if (threadIdx.x < o) sm[threadIdx.x] += sm[threadIdx.x + o];
    __syncthreads();
  }
  if (threadIdx.x == 0) MEAN[b] = sm[0] * (1.0f / ((float)NXg * (float)NXg));
}

__global__ void dalpha_kernel(const float* __restrict__ MEAN, const float* __restrict__ alpha,
                              float* __restrict__ DA) {
  int tid = threadIdx.x;
  if (tid < 8) {
    int e = tid >> 1, i = tid & 1;
    DA[tid] = alpha[i * 2 + 0] * MEAN[e * 2 + 0] + alpha[i * 2 + 1] * MEAN[e * 2 + 1];
  }
}

// mode->layer transform fused with homogeneous-solution correction + unpack
__global__ void dp_kernel(const float* __restrict__ GA, const float* __restrict__ Cm2l,
                          const float* __restrict__ DA, const float* __restrict__ hom,
                          float* __restrict__ dp) {
  int e = blockIdx.z;
  int x = blockIdx.y * 16 + threadIdx.y;
  int y = blockIdx.x * 16 + threadIdx.x;
  if (x >= NXg || y >= NXg) return;
  size_t pidx = (size_t)x * NXg + y;
  bool interior = (x >= 1 && x <= NIg && y >= 1 && y <= NIg);
  float mv[3];
#pragma unroll
  for (int j = 0; j < 3; ++j) {
    float v = 0.0f;
    if (interior) v = GA[((size_t)(e * Lg + j)) * NPNP + (size_t)(x - 1) * NPg + (y - 1)];
    if (j < 2) v += DA[e * 2 + j] * hom[(size_t)j * FIELD + pidx];
    mv[j] = v;
  }
#pragma unroll
  for (int i = 0; i < 3; ++i) {
    dp[((size_t)(e * Lg + i)) * FIELD + pidx] =
        Cm2l[i * 3 + 0] * mv[0] + Cm2l[i * 3 + 1] * mv[1] + Cm2l[i * 3 + 2] * mv[2];
  }
}

// dq boundary update: dq_bound = lap_bnd(dp/(F0*DX)^2) - A @ dp_bound
__global__ void boundary_kernel(const float* __restrict__ dp, const float* __restrict__ Amat,
                                float* __restrict__ dq) {
  int idx = blockIdx.x * 256 + threadIdx.x;   // 0..2047
  int e = blockIdx.y;
  if (idx >= 2 * NIg + 2 * NXg) return;
  int xe, ye, xi, yi;
  if (idx < NIg)                { xe = 0;              ye = 1 + idx;             xi = 1;       yi = ye; }
  else if (idx < 2 * NIg)       { xe = NXg - 1;        ye = 1 + (idx - NIg);     xi = NXg - 2; yi = ye; }
  else if (idx < 2 * NIg + NXg) { xe = idx - 2 * NIg;  ye = 0;                   xi = xe;      yi = 1; }
  else                          { xe = idx - (2 * NIg + NXg); ye = NXg - 1;      xi = xe;      yi = NXg - 2; }

  float dpv[3], dlap[3];
#pragma unroll
  for (int l = 0; l < 3; ++l) {
    float edge  = dp[((size_t)(e * Lg + l)) * FIELD + (size_t)xe * NXg + ye];
    float inner = dp[((size_t)(e * Lg + l)) * FIELD + (size_t)xi * NXg + yi];
    dpv[l]  = edge;
    dlap[l] = BNDCc * (inner - edge);
  }
#pragma unroll
  for (int i = 0; i < 3; ++i) {
    float v = dlap[i] - (Amat[i * 3 + 0] * dpv[0] + Amat[i * 3 + 1] * dpv[1] +
                         Amat[i * 3 + 2] * dpv[2]);
    dq[((size_t)(e * Lg + i)) * FIELD + (size_t)xe * NXg + ye] = v;
  }
}

// yt = y + c*dt*k   (dt read on-device from t[] to stay capture-safe)
__global__ void axpy_kernel(const float* __restrict__ y, const float* __restrict__ k,
                            const float* __restrict__ t, int step, float c,
                            float* __restrict__ yt) {
  size_t i = (size_t)blockIdx.x * 256 + threadIdx.x;
  if (i < STATE) {
    float dt = t[step + 1] - t[step];
    yt[i] = y[i] + c * dt * k[i];
  }
}

// y = y + dt/6*(k1 + 2k2 + 2k3 + k4), also stored to trajectory slab
__global__ void rk4_combine_kernel(float* __restrict__ y, const float* __restrict__ k1,
                                   const float* __restrict__ k2, const float* __restrict__ k3,
                                   const float* __restrict__ k4, const float* __restrict__ t,
                                   int step, float* __restrict__ out) {
  size_t i = (size_t)blockIdx.x * 256 + threadIdx.x;
  if (i < STATE) {
    float dt6 = (t[step + 1] - t[step]) * (1.0f / 6.0f);
    float v = y[i] + dt6 * (k1[i] + 2.0f * (k2[i] + k3[i]) + k4[i]);
    y[i] = v;
    out[i] = v;
  }
}

// ---------------------------------------------------------------------------
// host-side derivative enqueue
// ---------------------------------------------------------------------------
static void enqueue_deriv(const float* y, float* k,
                          const float* Amat, const float* Cl2m, const float* Cm2l,
                          const float* helm, const float* alpha, const float* hom,
                          const float* wind,
                          float* D2, float* LP2, float* GA, float* GB, float* S,
                          float* MEAN, float* DA, hipStream_t stream) {
  dim3 b16(16, 16);
  dim3 gF((NXg + 15) / 16, (NXg + 15) / 16, Eg * Lg);
  dim3 gFe((NXg + 15) / 16, (NXg + 15) / 16, Eg);
  dim3 gP(NPg / 16, NPg / 16, Eg);
  dim3 gP12(NPg / 16, NPg / 16, Eg * Lg);
  dim3 gG(NPg / 64, NPg / 64, Eg * Lg);

  // delta2_p and lap(delta2_p)
  lap_bc_kernel<<<gF, b16, 0, stream>>>(y + ELXY, D2);
  lap_bc_kernel<<<gF, b16, 0, stream>>>(D2, LP2);
  // interior rhs -> dq
  rhs_kernel<<<gF, b16, 0, stream>>>(y, LP2, wind, k);
  // layer->mode + pack
  cl2m_pack_kernel<<<gP, b16, 0, stream>>>(k, Cl2m, GA);
  // inverse elliptic: S*((S*f*S)/helm)*S as 4 batched WMMA GEMMs
  gemm512_wmma<<<gG, 256, 0, stream>>>(S, 0, GA, (long long)NPNP, GB);
  gemm512_wmma<<<gG, 256, 0, stream>>>(GB, (long long)NPNP, S, 0, GA);
  helm_scale_kernel<<<gP12, b16, 0, stream>>>(GA, helm);
  gemm512_wmma<<<gG, 256, 0, stream>>>(S, 0, GA, (long long)NPNP, GB);
  gemm512_wmma<<<gG, 256, 0, stream>>>(GB, (long long)NPNP, S, 0, GA);
  // mass-conservation correction
  mean_reduce_kernel<<<Eg * 2, 256, 0, stream>>>(GA, MEAN);
  dalpha_kernel<<<1, 32, 0, stream>>>(MEAN, alpha, DA);
  // mode->layer -> dp
  dp_kernel<<<gFe, b16, 0, stream>>>(GA, Cm2l, DA, hom, k + ELXY);
  // dq boundary conditions
  boundary_kernel<<<dim3(8, Eg), 256, 0, stream>>>(k + ELXY, Amat, k);
}

extern "C" void kernel_launch(void* const* d_in, const int* in_sizes, int n_in,
                              void* d_out, int out_size, void* d_ws, size_t ws_size,
                              hipStream_t stream) {
  (void)in_sizes; (void)n_in; (void)out_size; (void)ws_size;
  const float* y0    = (const float*)d_in[0];
  const float* tarr  = (const float*)d_in[1];
  const float* Amat  = (const float*)d_in[2];
  const float* Cl2m  = (const float*)d_in[3];
  const float* Cm2l  = (const float*)d_in[4];
  const float* helm  = (const float*)d_in[5];
  const float* alpha = (const float*)d_in[6];
  const float* hom   = (const float*)d_in[7];
  const float* wind  = (const float*)d_in[8];
  float* out = (float*)d_out;
  float* ws  = (float*)d_ws;

  // workspace layout (floats)
  float* Y    = ws;
  float* YT   = Y  + STATE;
  float* K1   = YT + STATE;
  float* K2   = K1 + STATE;
  float* K3   = K2 + STATE;
  float* K4   = K3 + STATE;
  float* D2   = K4 + STATE;
  float* LP2  = D2 + ELXY;
  float* GA   = LP2 + ELXY;
  float* GB   = GA + GSZ;
  float* S    = GB + GSZ;
  float* MEAN = S  + NPNP;
  float* DA   = MEAN + 8;

  // DST-I matrix (deterministic, rebuilt each call)
  make_S_kernel<<<dim3(NPg / 16, NPg / 16), dim3(16, 16), 0, stream>>>(S);

  // trajectory slab 0 = y0; working state = y0
  hipMemcpyAsync((void*)Y,   (const void*)y0, STATE * sizeof(float),
                 hipMemcpyDeviceToDevice, stream);
  hipMemcpyAsync((void*)out, (const void*)y0, STATE * sizeof(float),
                 hipMemcpyDeviceToDevice, stream);

  const size_t nBlk = (STATE + 255) / 256;
  for (int s = 0; s < 4; ++s) {
    enqueue_deriv(Y,  K1, Amat, Cl2m, Cm2l, helm, alpha, hom, wind,
                  D2, LP2, GA, GB, S, MEAN, DA, stream);
    axpy_kernel<<<nBlk, 256, 0, stream>>>(Y, K1, tarr, s, 0.5f, YT);
    enqueue_deriv(YT, K2, Amat, Cl2m, Cm2l, helm, alpha, hom, wind,
                  D2, LP2, GA, GB, S, MEAN, DA, stream);
    axpy_kernel<<<nBlk, 256, 0, stream>>>(Y, K2, tarr, s, 0.5f, YT);
    enqueue_deriv(YT, K3, Amat, Cl2m, Cm2l, helm, alpha, hom, wind,
                  D2, LP2, GA, GB, S, MEAN, DA, stream);
    axpy_kernel<<<nBlk, 256, 0, stream>>>(Y, K3, tarr, s, 1.0f, YT);
    enqueue_deriv(YT, K4, Amat, Cl2m, Cm2l, helm, alpha, hom, wind,
                  D2, LP2, GA, GB, S, MEAN, DA, stream);
    rk4_combine_kernel<<<nBlk, 256, 0, stream>>>(Y, K1, K2, K3, K4, tarr, s,
                                                 out + (size_t)(s + 1) * STATE);
  }
}